// NT_Xent_24361054503016
// MI455X (gfx1250) — compile-verified
//
#include <hip/hip_runtime.h>
#include <math.h>

#define D        512
#define NROWS    8192        // N = 2*BATCH
#define BATCH    4096
#define CSPLIT   8           // column chunks (grid.y)
#define TN       32          // cols per tile (two 16-wide WMMA frags)
#define TILES_PER_CHUNK (NROWS / TN / CSPLIT)   // 32
#define ENC_NEG_INF 0x007FFFFFu                  // ordered-uint encoding of -inf

// Chunked k-major LDS layout (zero padding, bank-conflict-free frag loads):
//   As[k4][128 rows][4]: 128*512 = 65536 floats (256 KB)
//   Bs[k4][32 cols][4] : 128*128 = 16384 floats ( 64 KB)   -> total exactly 320 KB
#define AS_K4  512           // floats per k4-chunk of As
#define BS_K4  128           // floats per k4-chunk of Bs

typedef float v2f __attribute__((ext_vector_type(2)));
typedef float v8f __attribute__((ext_vector_type(8)));

__device__ __forceinline__ unsigned enc_f32(float f) {
    unsigned b = __float_as_uint(f);
    return (b & 0x80000000u) ? ~b : (b | 0x80000000u);
}
__device__ __forceinline__ float dec_f32(unsigned u) {
    return (u & 0x80000000u) ? __uint_as_float(u ^ 0x80000000u) : __uint_as_float(~u);
}

// ---------------- K0: inverse row norms (msg rows 0..8191, img rows 8192..16383)
__global__ void inv_norm_kernel(const float* __restrict__ msg,
                                const float* __restrict__ img,
                                float* __restrict__ invAll) {
    int lane = threadIdx.x & 31;
    int wv   = threadIdx.x >> 5;
    int row  = blockIdx.x * 8 + wv;          // 2048 blocks * 8 waves = 16384 rows
    const float* src = (row < NROWS) ? (msg + (size_t)row * D)
                                     : (img + (size_t)(row - NROWS) * D);
    float s = 0.f;
#pragma unroll
    for (int u = 0; u < 4; ++u) {
        float4 v = *(const float4*)(src + u * 128 + lane * 4);
        s += v.x * v.x + v.y * v.y + v.z * v.z + v.w * v.w;
    }
#pragma unroll
    for (int m = 1; m <= 16; m <<= 1) s += __shfl_xor(s, m, 32);
    if (lane == 0) invAll[row] = 1.0f / fmaxf(sqrtf(s), 1e-8f);
}

// ---------------- K1: init per-row accumulators
__global__ void zero_ws_kernel(float* rowSum, float* rowPos, unsigned* rowMaxBits,
                               float* accum) {
    int i = blockIdx.x * 256 + threadIdx.x;   // 32 blocks * 256 = 8192
    rowSum[i] = 0.f;
    rowPos[i] = 0.f;
    rowMaxBits[i] = ENC_NEG_INF;
    if (i < 2) accum[i] = 0.f;
}

// ---------------- K2: fused fp32-WMMA GEMM + online exp-sum / max / pos
__global__ __launch_bounds__(256, 1)
void ntxent_main_kernel(const float* __restrict__ msg,
                        const float* __restrict__ img,
                        const float* __restrict__ invAll,
                        float* __restrict__ rowSum,
                        float* __restrict__ rowPos,
                        unsigned* __restrict__ rowMaxBits) {
    extern __shared__ float smem[];
    float* As = smem;            // 65536 floats
    float* Bs = smem + 65536;    // 16384 floats

    const int tid   = threadIdx.x;
    const int lane  = tid & 31;
    const int wv    = tid >> 5;        // 8 waves
    const int hlf   = lane >> 4;       // half-wave
    const int nlane = lane & 15;
    const int rowBase  = blockIdx.x * 128;
    const int colChunk = blockIdx.y;   // 0..CSPLIT-1

    // ---- stage the 128x512 A strip once via async global->LDS DMA into chunked layout
    {
        int r  = tid >> 1;
        int k0 = (tid & 1) * 256;
        unsigned gbase = (((unsigned)(rowBase + r)) * D + (unsigned)k0) * 4u;
        unsigned lbase = (unsigned)(uintptr_t)smem
                       + (((unsigned)(k0 >> 2)) * AS_K4 + (unsigned)r * 4u) * 4u;
#pragma unroll
        for (int u = 0; u < 64; ++u) {
            unsigned loff = lbase + (unsigned)(AS_K4 * 4) * (unsigned)u;  // +2048 B per k4
            unsigned goff = gbase + 16u * (unsigned)u;
            asm volatile("global_load_async_to_lds_b128 %0, %1, %2"
                         :: "v"(loff), "v"(goff), "s"(msg) : "memory");
        }
    }
    asm volatile("s_wait_asynccnt 0x0" ::: "memory");
    __syncthreads();

    // C-layout row for accumulator element v: rowC0 + v  (lanes 16-31 -> +8)
    const int rowC0 = rowBase + wv * 16 + hlf * 8;
    float invA10[8];
#pragma unroll
    for (int v = 0; v < 8; ++v) invA10[v] = invAll[rowC0 + v] * 10.0f;  // fold 1/TEMP

    float sum[8], pos[8], mx[8];
#pragma unroll
    for (int v = 0; v < 8; ++v) { sum[v] = 0.f; pos[v] = 0.f; mx[v] = -INFINITY; }

    // Fragment base offsets in the chunked layouts (float units)
    const int aoff  = (wv * 16 + nlane) * 4 + 2 * hlf;   // + k4*512
    const int boff0 = nlane * 4 + 2 * hlf;               // + k4*128   (cols 0..15)

    // B staging map: thread -> (col n, 16 consecutive k4 chunks), reg double-buffered
    const int bn  = tid & 31;                 // col within tile
    const int bk4 = (tid >> 5) * 16;          // first k4 chunk (16 per thread)

    const int tileBegin = colChunk * TILES_PER_CHUNK;
    float4 breg[16];
    {   // prologue: fetch first tile into registers
        const float4* src = (const float4*)(img + (size_t)(tileBegin * TN + bn) * D + bk4 * 4);
#pragma unroll
        for (int u = 0; u < 16; ++u) breg[u] = src[u];
    }

    for (int t = 0; t < TILES_PER_CHUNK; ++t) {
        const int c0 = (tileBegin + t) * TN;
        __syncthreads();   // previous tile's Bs consumers done
        {   // commit registers -> LDS chunked layout
#pragma unroll
            for (int u = 0; u < 16; ++u)
                *(float4*)(Bs + (bk4 + u) * BS_K4 + bn * 4) = breg[u];
        }
        __syncthreads();
        if (t + 1 < TILES_PER_CHUNK) {   // uniform branch; latency hidden by K-loop below
            const float4* src = (const float4*)(img + (size_t)(c0 + TN + bn) * D + bk4 * 4);
#pragma unroll
            for (int u = 0; u < 16; ++u) breg[u] = src[u];
        }

        // per-tile column inverse norms: issue early, consumed after the K-loop
        const float invB0 = invAll[NROWS + c0 + nlane];
        const float invB1 = invAll[NROWS + c0 + 16 + nlane];

        v8f cA = {};
        v8f cB = {};
        {   // K-loop: 16 x (unrolled 8) with immediate-offset DS addressing
            const float* pA = As + aoff;
            const float* pB = Bs + boff0;
            for (int kb = 0; kb < 16; ++kb) {
#pragma unroll
                for (int u = 0; u < 8; ++u) {
                    v2f a  = *(const v2f*)(pA + u * AS_K4);
                    v2f b0 = *(const v2f*)(pB + u * BS_K4);
                    v2f b1 = *(const v2f*)(pB + u * BS_K4 + 64);
                    cA = __builtin_amdgcn_wmma_f32_16x16x4_f32(
                             false, a, false, b0, (short)0, cA, false, false);
                    cB = __builtin_amdgcn_wmma_f32_16x16x4_f32(
                             false, a, false, b1, (short)0, cB, false, false);
                }
                pA += 8 * AS_K4;
                pB += 8 * BS_K4;
            }
        }

#pragma unroll
        for (int q = 0; q < 2; ++q) {
            const int j = c0 + q * 16 + nlane;
            const float invB = q ? invB1 : invB0;
#pragma unroll
            for (int v = 0; v < 8; ++v) {
                const int i = rowC0 + v;
                const float cv = q ? cB[v] : cA[v];
                const float s = cv * invA10[v] * invB;         // cos/TEMP
                const float e = __expf(s);
                const bool diag = (j == i);
                const bool part = (j == (i ^ BATCH));
                sum[v] += diag ? 0.f : e;                      // LSE includes the positive
                pos[v] += part ? s : 0.f;                      // exactly one hit per row
                mx[v]   = fmaxf(mx[v], (diag || part) ? -INFINITY : s);
            }
        }
    }

    // reduce across the 16 lanes that share each row
#pragma unroll
    for (int m = 1; m <= 8; m <<= 1) {
#pragma unroll
        for (int v = 0; v < 8; ++v) {
            sum[v] += __shfl_xor(sum[v], m, 32);
            pos[v] += __shfl_xor(pos[v], m, 32);
            mx[v]   = fmaxf(mx[v], __shfl_xor(mx[v], m, 32));
        }
    }
    if (nlane == 0) {
#pragma unroll
        for (int v = 0; v < 8; ++v) {
            atomicAdd(rowSum + rowC0 + v, sum[v]);
            atomicAdd(rowPos + rowC0 + v, pos[v]);
            atomicMax(rowMaxBits + rowC0 + v, enc_f32(mx[v]));
        }
    }
}

// ---------------- K3: per-row loss/acc, reduce to two scalars
__global__ void row_final_kernel(const float* __restrict__ rowSum,
                                 const float* __restrict__ rowPos,
                                 const unsigned* __restrict__ rowMaxBits,
                                 float* __restrict__ accum) {
    int i = blockIdx.x * 256 + threadIdx.x;   // 8192 rows
    float p    = rowPos[i];
    float loss = logf(rowSum[i]) - p;
    float acc  = (p >= dec_f32(rowMaxBits[i])) ? 1.f : 0.f;
#pragma unroll
    for (int m = 1; m <= 16; m <<= 1) {
        loss += __shfl_xor(loss, m, 32);
        acc  += __shfl_xor(acc, m, 32);
    }
    if ((threadIdx.x & 31) == 0) {
        atomicAdd(accum + 0, loss);
        atomicAdd(accum + 1, acc);
    }
}

// ---------------- K4: finalize
__global__ void finalize_kernel(const float* __restrict__ accum, float* __restrict__ out) {
    out[0] = accum[0] / (float)NROWS;
    out[1] = accum[1] / (float)NROWS;
}

extern "C" void kernel_launch(void* const* d_in, const int* in_sizes, int n_in,
                              void* d_out, int out_size, void* d_ws, size_t ws_size,
                              hipStream_t stream) {
    const float* msg = (const float*)d_in[0];
    const float* img = (const float*)d_in[1];
    float* out = (float*)d_out;

    float* ws = (float*)d_ws;
    float*    invAll     = ws;                       // 16384
    float*    rowSum     = ws + 16384;               // 8192
    float*    rowPos     = ws + 24576;               // 8192
    unsigned* rowMaxBits = (unsigned*)(ws + 32768);  // 8192
    float*    accum      = ws + 40960;               // 2

    inv_norm_kernel<<<2048, 256, 0, stream>>>(msg, img, invAll);
    zero_ws_kernel<<<32, 256, 0, stream>>>(rowSum, rowPos, rowMaxBits, accum);

    const size_t ldsBytes = (size_t)(65536 + 16384) * sizeof(float); // 327,680 B = 320 KB
    dim3 grid(NROWS / 128, CSPLIT);
    ntxent_main_kernel<<<grid, 256, ldsBytes, stream>>>(
        msg, img, invAll, rowSum, rowPos, rowMaxBits);

    row_final_kernel<<<32, 256, 0, stream>>>(rowSum, rowPos, rowMaxBits, accum);
    finalize_kernel<<<1, 1, 0, stream>>>(accum, out);
}